// VectorQuantizer_47674136986185
// MI455X (gfx1250) — compile-verified
//
#include <hip/hip_runtime.h>
#include <stdint.h>

typedef __attribute__((ext_vector_type(2))) float        v2f;
typedef __attribute__((ext_vector_type(4))) float        v4f;
typedef __attribute__((ext_vector_type(8))) float        v8f;
typedef __attribute__((ext_vector_type(4))) unsigned int v4u;
typedef __attribute__((ext_vector_type(4))) int          v4i;
typedef __attribute__((ext_vector_type(8))) int          v8i;

#define DDIM        512
#define KEMB        8192
#define NROWS       4096
#define NTILES      (KEMB / 16)      // 512 embedding tiles of 16
#define TILE_FLOATS (16 * DDIM)      // 8192 floats per swizzled tile (32 KB)

#if __has_builtin(__builtin_amdgcn_tensor_load_to_lds)
#define USE_TDM 1
#endif

// ---------------------------------------------------------------------------
// Kernel 0: init packed min-keys to all-ones
// ---------------------------------------------------------------------------
__global__ void vq_init_kernel(unsigned long long* __restrict__ minkey) {
    int i = blockIdx.x * blockDim.x + threadIdx.x;
    if (i < NROWS) minkey[i] = ~0ULL;
}

// ---------------------------------------------------------------------------
// Kernel 1: eNorm[k] = ||embedding[k]||^2   (one wave32 per codebook row)
// ---------------------------------------------------------------------------
__global__ __launch_bounds__(256)
void vq_enorm_kernel(const float* __restrict__ emb, float* __restrict__ eNorm) {
    const int wave = threadIdx.x >> 5;
    const int lane = threadIdx.x & 31;
    const int row  = blockIdx.x * 8 + wave;
    const float* p = emb + (size_t)row * DDIM;
    float s = 0.0f;
#pragma unroll
    for (int t = 0; t < DDIM / 32; ++t) {
        float x = p[lane + 32 * t];
        s += x * x;
    }
#pragma unroll
    for (int off = 16; off > 0; off >>= 1) s += __shfl_xor(s, off, 32);
    if (lane == 0) eNorm[row] = s;
}

// ---------------------------------------------------------------------------
// Kernel 1b: swizzle embedding into WMMA B-fragment order:
//   embSw[tile][kp][lane][e], e=0..3 -> (kk = 2*kp + (e>>1), elem = e&1)
//   B-frag float for lane l=half*16+ln, kk: emb[tile*16+ln][4*kk + 2*half + elem]
// Per-wave loads of B become contiguous 512B global_load_b128 streams.
// ---------------------------------------------------------------------------
__global__ __launch_bounds__(256)
void vq_bswizzle_kernel(const float* __restrict__ emb, float* __restrict__ embSw) {
    int idx = blockIdx.x * 256 + threadIdx.x;        // 0 .. KEMB*DDIM-1
    int t   = idx >> 13;                             // / TILE_FLOATS
    int rem = idx & (TILE_FLOATS - 1);
    int kp  = rem >> 7;
    int l   = (rem >> 2) & 31;
    int e   = rem & 3;
    int hf  = l >> 4;
    int ln  = l & 15;
    int c   = 8 * kp + 4 * (e >> 1) + 2 * hf + (e & 1);
    embSw[idx] = emb[(size_t)(t * 16 + ln) * DDIM + c];
}

// ---------------------------------------------------------------------------
// Kernel 2: fp32 WMMA GEMM (scores = eNorm - 2 x.e) + argmin via u64 atomicMin
// One block = one 16-row M tile; 16 waves each own a 512-wide K chunk,
// processing 4 output tiles per pass (one A fragment feeds 8 WMMAs).
// ---------------------------------------------------------------------------
__device__ __forceinline__ unsigned int float_orderkey(float f) {
    unsigned int b = __float_as_uint(f);
    return (b & 0x80000000u) ? ~b : (b | 0x80000000u);   // monotone f -> u32
}

__global__ __launch_bounds__(512)
void vq_argmin_kernel(const float* __restrict__ lat,
                      const float* __restrict__ embSw,
                      const float* __restrict__ eNorm,
                      unsigned long long* __restrict__ minkey) {
    __shared__ float As[TILE_FLOATS];        // swizzled A: [kp][lane][4]
#ifdef USE_TDM
    __shared__ float Araw[16 * DDIM];        // TDM-staged raw A tile
#endif
    const int m0   = blockIdx.x * 16;
    const int tid  = threadIdx.x;
    const int wave = tid >> 5;
    const int lane = tid & 31;

#ifdef USE_TDM
    // --- Tensor Data Mover: 2D tile 16 rows x 512 cols fp32, stride 512 ---
    if (wave == 0) {
        unsigned long long ga = (unsigned long long)(uintptr_t)(lat + (size_t)m0 * DDIM);
        unsigned int ldsa = (unsigned int)(uintptr_t)(&Araw[0]);
        v4u g0;
        g0[0] = 1u;                                        // count=1, user mode
        g0[1] = ldsa;                                      // lds_addr
        g0[2] = (unsigned int)ga;                          // global_addr[31:0]
        g0[3] = (unsigned int)(ga >> 32) | (2u << 30);     // global_addr[56:32] | type=2
        v8i g1;
        g1[0] = (int)(2u << 16);          // workgroup_mask=0, data_size=2 (4B)
        g1[1] = (int)(DDIM << 16);        // tensor_dim0[15:0] = 512
        g1[2] = (int)(NROWS << 16);       // tensor_dim0 msbs=0 | tensor_dim1[15:0]
        g1[3] = (int)(DDIM << 16);        // tensor_dim1 msbs=0 | tile_dim0 = 512
        g1[4] = 16;                       // tile_dim1 = 16, tile_dim2 = 0
        g1[5] = DDIM;                     // tensor_dim0_stride[31:0] = 512
        g1[6] = 0;                        // stride msbs, tensor_dim1_stride lo
        g1[7] = 0;
        v4i gz4 = {0, 0, 0, 0};
        v8i gz8 = {0, 0, 0, 0, 0, 0, 0, 0};
        // clang-23 / therock toolchain: 6-arg form (extra int32x8 group, zero)
        __builtin_amdgcn_tensor_load_to_lds(g0, g1, gz4, gz4, gz8, 0);
        __builtin_amdgcn_s_wait_tensorcnt(0);
    }
    __syncthreads();
    // swizzle raw tile -> fragment order (one-time, 32 KB)
#pragma unroll
    for (int i = 0; i < 16; ++i) {
        int idx = tid + i * 512;
        int e = idx & 3, l = (idx >> 2) & 31, kp = idx >> 7;
        int hf = l >> 4, ln = l & 15;
        int c = 8 * kp + 4 * (e >> 1) + 2 * hf + (e & 1);
        As[idx] = Araw[ln * DDIM + c];
    }
#else
#pragma unroll
    for (int i = 0; i < 16; ++i) {
        int idx = tid + i * 512;
        int e = idx & 3, l = (idx >> 2) & 31, kp = idx >> 7;
        int hf = l >> 4, ln = l & 15;
        int c = 8 * kp + 4 * (e >> 1) + 2 * hf + (e & 1);
        As[idx] = lat[(size_t)(m0 + ln) * DDIM + c];
    }
#endif
    __syncthreads();

    const int half  = lane >> 4;
    const int ln    = lane & 15;
    const int tbase = wave * 32;     // first embedding tile of this wave's chunk

    float bv[8];
    int   bi[8];
#pragma unroll
    for (int j = 0; j < 8; ++j) { bv[j] = 3.4e38f; bi[j] = 0x7fffffff; }

    const float* aswl = As + lane * 4;

    for (int kt = 0; kt < 32; kt += 4) {
        const int t0 = tbase + kt;
        const float* p0 = embSw + (size_t)(t0 + 0) * TILE_FLOATS + lane * 4;
        const float* p1 = embSw + (size_t)(t0 + 1) * TILE_FLOATS + lane * 4;
        const float* p2 = embSw + (size_t)(t0 + 2) * TILE_FLOATS + lane * 4;
        const float* p3 = embSw + (size_t)(t0 + 3) * TILE_FLOATS + lane * 4;

        if (kt + 4 < 32) {   // uniform branch: prefetch next quad's streams
            __builtin_prefetch(p0 + 4 * TILE_FLOATS, 0, 1);
            __builtin_prefetch(p1 + 4 * TILE_FLOATS, 0, 1);
            __builtin_prefetch(p2 + 4 * TILE_FLOATS, 0, 1);
            __builtin_prefetch(p3 + 4 * TILE_FLOATS, 0, 1);
        }

        v8f c0 = {}, c1 = {}, c2 = {}, c3 = {};
#pragma unroll 4
        for (int kp = 0; kp < DDIM / 8; ++kp) {          // 64 iters, 2 kk each
            v4f a  = *(const v4f*)(aswl + kp * 128);     // ds_load_b128, no conflicts
            v4f b0 = *(const v4f*)(p0 + kp * 128);       // coalesced global_load_b128
            v4f b1 = *(const v4f*)(p1 + kp * 128);
            v4f b2 = *(const v4f*)(p2 + kp * 128);
            v4f b3 = *(const v4f*)(p3 + kp * 128);
            v2f alo = __builtin_shufflevector(a, a, 0, 1);
            v2f ahi = __builtin_shufflevector(a, a, 2, 3);
            c0 = __builtin_amdgcn_wmma_f32_16x16x4_f32(false, alo, false,
                     __builtin_shufflevector(b0, b0, 0, 1), (short)0, c0, false, false);
            c1 = __builtin_amdgcn_wmma_f32_16x16x4_f32(false, alo, false,
                     __builtin_shufflevector(b1, b1, 0, 1), (short)0, c1, false, false);
            c2 = __builtin_amdgcn_wmma_f32_16x16x4_f32(false, alo, false,
                     __builtin_shufflevector(b2, b2, 0, 1), (short)0, c2, false, false);
            c3 = __builtin_amdgcn_wmma_f32_16x16x4_f32(false, alo, false,
                     __builtin_shufflevector(b3, b3, 0, 1), (short)0, c3, false, false);
            c0 = __builtin_amdgcn_wmma_f32_16x16x4_f32(false, ahi, false,
                     __builtin_shufflevector(b0, b0, 2, 3), (short)0, c0, false, false);
            c1 = __builtin_amdgcn_wmma_f32_16x16x4_f32(false, ahi, false,
                     __builtin_shufflevector(b1, b1, 2, 3), (short)0, c1, false, false);
            c2 = __builtin_amdgcn_wmma_f32_16x16x4_f32(false, ahi, false,
                     __builtin_shufflevector(b2, b2, 2, 3), (short)0, c2, false, false);
            c3 = __builtin_amdgcn_wmma_f32_16x16x4_f32(false, ahi, false,
                     __builtin_shufflevector(b3, b3, 2, 3), (short)0, c3, false, false);
        }

        const int col0 = (t0 + 0) * 16 + ln;
        const int col1 = (t0 + 1) * 16 + ln;
        const int col2 = (t0 + 2) * 16 + ln;
        const int col3 = (t0 + 3) * 16 + ln;
        const float en0 = eNorm[col0];
        const float en1 = eNorm[col1];
        const float en2 = eNorm[col2];
        const float en3 = eNorm[col3];
#pragma unroll
        for (int j = 0; j < 8; ++j) {     // rows m0+j (half 0) / m0+j+8 (half 1)
            float s0 = en0 - 2.0f * c0[j];
            float s1 = en1 - 2.0f * c1[j];
            float s2 = en2 - 2.0f * c2[j];
            float s3 = en3 - 2.0f * c3[j];
            if (s0 < bv[j] || (s0 == bv[j] && col0 < bi[j])) { bv[j] = s0; bi[j] = col0; }
            if (s1 < bv[j] || (s1 == bv[j] && col1 < bi[j])) { bv[j] = s1; bi[j] = col1; }
            if (s2 < bv[j] || (s2 == bv[j] && col2 < bi[j])) { bv[j] = s2; bi[j] = col2; }
            if (s3 < bv[j] || (s3 == bv[j] && col3 < bi[j])) { bv[j] = s3; bi[j] = col3; }
        }
    }

    // Cross-lane (16-wide) argmin reduce, then one u64 atomicMin per row.
#pragma unroll
    for (int j = 0; j < 8; ++j) {
        float v = bv[j];
        int   i = bi[j];
#pragma unroll
        for (int off = 1; off < 16; off <<= 1) {
            float ov = __shfl_xor(v, off, 16);
            int   oi = __shfl_xor(i, off, 16);
            if (ov < v || (ov == v && oi < i)) { v = ov; i = oi; }
        }
        if (ln == 0) {
            unsigned long long key =
                ((unsigned long long)float_orderkey(v) << 32) | (unsigned int)i;
            atomicMin(&minkey[m0 + j + half * 8], key);
        }
    }
}

// ---------------------------------------------------------------------------
// Kernel 3: commitment loss (gather rows by target) + decode argmin indices
// ---------------------------------------------------------------------------
__global__ __launch_bounds__(256)
void vq_finalize_kernel(const float* __restrict__ lat,
                        const float* __restrict__ emb,
                        const int* __restrict__ target,
                        const unsigned long long* __restrict__ minkey,
                        float* __restrict__ out) {
    const int wave = threadIdx.x >> 5;
    const int lane = threadIdx.x & 31;
    const int row  = blockIdx.x * 8 + wave;
    const int tgt  = target[row];
    const float* e = emb + (size_t)tgt * DDIM;
    const float* x = lat + (size_t)row * DDIM;
    float s = 0.0f;
#pragma unroll
    for (int t = 0; t < DDIM / 32; ++t) {
        int d = lane + 32 * t;
        float df = e[d] - x[d];
        s += df * df;
    }
#pragma unroll
    for (int off = 16; off > 0; off >>= 1) s += __shfl_xor(s, off, 32);
    if (lane == 0) {
        out[row]         = s * (1.0f / (float)DDIM);
        out[NROWS + row] = (float)(unsigned int)(minkey[row] & 0xFFFFFFFFull);
    }
}

// ---------------------------------------------------------------------------
extern "C" void kernel_launch(void* const* d_in, const int* in_sizes, int n_in,
                              void* d_out, int out_size, void* d_ws, size_t ws_size,
                              hipStream_t stream) {
    const float* latents   = (const float*)d_in[0];
    const int*   target    = (const int*)d_in[1];
    const float* embedding = (const float*)d_in[2];
    float*       out       = (float*)d_out;

    // workspace layout: swizzled embedding (16 MB) | eNorm (32 KB) | minkey (32 KB)
    float*              embSw  = (float*)d_ws;
    float*              eNorm  = (float*)((char*)d_ws + (size_t)KEMB * DDIM * sizeof(float));
    unsigned long long* minkey = (unsigned long long*)((char*)eNorm + KEMB * sizeof(float));

    vq_init_kernel<<<(NROWS + 255) / 256, 256, 0, stream>>>(minkey);
    vq_enorm_kernel<<<KEMB / 8, 256, 0, stream>>>(embedding, eNorm);
    vq_bswizzle_kernel<<<(KEMB * DDIM) / 256, 256, 0, stream>>>(embedding, embSw);
    vq_argmin_kernel<<<NROWS / 16, 512, 0, stream>>>(latents, embSw, eNorm, minkey);
    vq_finalize_kernel<<<NROWS / 8, 256, 0, stream>>>(latents, embedding, target, minkey, out);
}